// MergeModel_48636209660176
// MI455X (gfx1250) — compile-verified
//
#include <hip/hip_runtime.h>

typedef __attribute__((ext_vector_type(2))) float v2f;
typedef __attribute__((ext_vector_type(8))) float v8f;

// ---------------------------------------------------------------------------
// Utility kernels
// ---------------------------------------------------------------------------

__global__ void zero_kernel(float* __restrict__ p, long n) {
    long i = (long)blockIdx.x * blockDim.x + threadIdx.x;
    if (i < n) p[i] = 0.0f;
}

// One block (256 threads) per edge: coalesced 1KB read of edge_attr row,
// scatter atomic-add into the agg half (cols 0..255) of h0[col].
__global__ __launch_bounds__(256) void scatter_add_kernel(
    const float* __restrict__ edge_attr, const int* __restrict__ col,
    float* __restrict__ h0, float* __restrict__ cnt) {
    int e = blockIdx.x;
    int f = threadIdx.x;
    int c = col[e];
    atomicAdd(&h0[(long)c * 512 + f], edge_attr[(long)e * 256 + f]);
    if (f == 0) atomicAdd(&cnt[c], 1.0f);
}

// Divide agg sums by max(count,1) and fill x into cols 256..511 of h0.
__global__ __launch_bounds__(256) void finalize_h0_kernel(
    const float* __restrict__ x, const float* __restrict__ cnt,
    float* __restrict__ h0) {
    int i = blockIdx.x;
    int f = threadIdx.x;
    float s = cnt[i];
    s = s > 1.0f ? s : 1.0f;
    long base = (long)i * 512;
    h0[base + f] = h0[base + f] / s;
    h0[base + 256 + f] = x[(long)i * 256 + f];
}

// In-place LayerNorm over D=2048, one block per row.
__global__ __launch_bounds__(256) void layernorm_kernel(
    float* __restrict__ h, const float* __restrict__ g,
    const float* __restrict__ b) {
    __shared__ float s_sum[256];
    __shared__ float s_sq[256];
    int i = blockIdx.x, t = threadIdx.x;
    float* row = h + (long)i * 2048;
    float v[8];
    float sum = 0.0f, sq = 0.0f;
#pragma unroll
    for (int j = 0; j < 8; ++j) {
        v[j] = row[t + 256 * j];
        sum += v[j];
        sq += v[j] * v[j];
    }
    s_sum[t] = sum;
    s_sq[t] = sq;
    __syncthreads();
    for (int off = 128; off > 0; off >>= 1) {
        if (t < off) { s_sum[t] += s_sum[t + off]; s_sq[t] += s_sq[t + off]; }
        __syncthreads();
    }
    float mu = s_sum[0] * (1.0f / 2048.0f);
    float var = s_sq[0] * (1.0f / 2048.0f) - mu * mu;
    float rs = rsqrtf(var + 1e-5f);
#pragma unroll
    for (int j = 0; j < 8; ++j) {
        int cidx = t + 256 * j;
        row[cidx] = (v[j] - mu) * rs * g[cidx] + b[cidx];
    }
}

// ---------------------------------------------------------------------------
// FP32 WMMA GEMM: C[M,N] = op(A[M,K] @ B[K,N] (+bias) (+C) (relu))
// All strides/dims are template constants so every inner-loop load folds
// k*stride into the 24-bit instruction immediate offset (no per-iteration
// 64-bit VALU address chains).
// Each wave computes one 32x64 C tile: 8 x v8f accumulators, 8 WMMAs per
// k-step of V_WMMA_F32_16X16X4_F32 against 2 A-frag loads (b64) and 8 B
// scalar loads. Waves in a block share tn -> B fragments are L0-resident.
// M must be a multiple of 32 (20000 = 625*32); N of 64; K of 4.
// ---------------------------------------------------------------------------
template <int LDA, int N, int K, bool BIAS, bool RELU, bool ACC>
__global__ __launch_bounds__(256) void wmma_gemm_f32(
    const float* __restrict__ A, const float* __restrict__ B,
    float* __restrict__ C, const float* __restrict__ bias, int tilesM) {
    constexpr int tilesN = N / 64;
    int wave = threadIdx.x >> 5;
    int lane = threadIdx.x & 31;
    long tile = (long)blockIdx.x * 8 + wave;
    long total = (long)tilesM * tilesN;
    if (tile >= total) return;  // wave-uniform: EXEC stays all-1s for WMMA
    int tm = (int)(tile % tilesM);
    int tn = (int)(tile / tilesM);
    int r0 = tm * 32, c0 = tn * 64;
    int half = lane >> 4, lm = lane & 15;

    v8f acc[2][4];
#pragma unroll
    for (int i = 0; i < 2; ++i)
#pragma unroll
        for (int j = 0; j < 4; ++j) acc[i][j] = (v8f){};

    // A fragment (16x4 fp32): lanes 0-15 hold K,K+1; lanes 16-31 hold K+2,K+3
    const float* Arow0 = A + (long)(r0 + lm) * LDA + 2 * half;
    const float* Arow1 = Arow0 + 16 * LDA;
    // B fragment (4x16 fp32): row k+2*half in .x, row k+2*half+1 in .y
    const float* Bbase = B + (long)(2 * half) * N + c0 + lm;

#pragma unroll 4
    for (int k = 0; k < K; k += 4) {
        v2f a0 = *(const v2f*)(Arow0 + k);
        v2f a1 = *(const v2f*)(Arow1 + k);
        v2f bf[4];
#pragma unroll
        for (int j = 0; j < 4; ++j) {
            bf[j].x = Bbase[(long)k * N + 16 * j];
            bf[j].y = Bbase[(long)k * N + N + 16 * j];
        }
#pragma unroll
        for (int j = 0; j < 4; ++j) {
            acc[0][j] = __builtin_amdgcn_wmma_f32_16x16x4_f32(
                false, a0, false, bf[j], (short)0, acc[0][j], false, false);
            acc[1][j] = __builtin_amdgcn_wmma_f32_16x16x4_f32(
                false, a1, false, bf[j], (short)0, acc[1][j], false, false);
        }
    }

    // Epilogue per 16x16 f32 C/D layout:
    // VGPR v, lanes 0-15: M=v, N=lane ; lanes 16-31: M=v+8, N=lane-16
    int cA = c0 + lm;
    float bv[4];
#pragma unroll
    for (int j = 0; j < 4; ++j) bv[j] = BIAS ? bias[cA + 16 * j] : 0.0f;
#pragma unroll
    for (int v = 0; v < 8; ++v) {
        int rA = r0 + 8 * half + v;
        float* p0 = C + (long)rA * N;
        float* p1 = p0 + 16 * N;
#pragma unroll
        for (int j = 0; j < 4; ++j) {
            int cc = cA + 16 * j;
            float o0 = acc[0][j][v];
            float o1 = acc[1][j][v];
            if (BIAS) { o0 += bv[j]; o1 += bv[j]; }
            if (ACC) { o0 += p0[cc]; o1 += p1[cc]; }
            if (RELU) { o0 = fmaxf(o0, 0.0f); o1 = fmaxf(o1, 0.0f); }
            p0[cc] = o0;
            p1[cc] = o1;
        }
    }
}

// ---------------------------------------------------------------------------
// Host launcher
// ---------------------------------------------------------------------------
extern "C" void kernel_launch(void* const* d_in, const int* in_sizes, int n_in,
                              void* d_out, int out_size, void* d_ws, size_t ws_size,
                              hipStream_t stream) {
    (void)n_in; (void)out_size; (void)ws_size;
    const float* x   = (const float*)d_in[0];
    const int*   ei  = (const int*)d_in[1];
    const float* ea  = (const float*)d_in[2];
    const float* W1  = (const float*)d_in[6];
    const float* b1  = (const float*)d_in[7];
    const float* W3  = (const float*)d_in[8];
    const float* b3  = (const float*)d_in[9];
    const float* lng = (const float*)d_in[10];
    const float* lnb = (const float*)d_in[11];
    const float* W6  = (const float*)d_in[12];
    const float* b6  = (const float*)d_in[13];
    const float* W8  = (const float*)d_in[14];
    const float* b8  = (const float*)d_in[15];

    const int N = in_sizes[0] / 256;  // 20000
    const int E = in_sizes[2] / 256;  // 640000
    const int* col = ei + E;          // edge_index[1]

    // Workspace layout: h0 [N,512] | cnt [N] | B1 [N,2048] | B2 [N,2048]
    float* h0  = (float*)d_ws;
    float* cnt = h0 + (long)N * 512;
    float* B1  = cnt + N;
    float* B2  = B1 + (long)N * 2048;

    // 1) zero h0 + cnt (contiguous), scatter-mean, build h0 = [agg | x]
    long zn = (long)N * 513;
    zero_kernel<<<(int)((zn + 255) / 256), 256, 0, stream>>>(h0, zn);
    scatter_add_kernel<<<E, 256, 0, stream>>>(ea, col, h0, cnt);
    finalize_h0_kernel<<<N, 256, 0, stream>>>(x, cnt, h0);

    const int tilesM = N / 32;  // 625
    auto blocks = [](long tm, long tn) { return (int)((tm * tn + 7) / 8); };

    // 2) h2 = relu(h0 @ W1 + b1)       [N,512]x[512,2048] -> B1
    wmma_gemm_f32<512, 2048, 512, true, true, false>
        <<<blocks(tilesM, 32), 256, 0, stream>>>(h0, W1, B1, b1, tilesM);

    // 3) h4 = relu(h2 @ W3 + b3)       [N,2048]x[2048,2048] -> B2
    wmma_gemm_f32<2048, 2048, 2048, true, true, false>
        <<<blocks(tilesM, 32), 256, 0, stream>>>(B1, W3, B2, b3, tilesM);

    // 4) h5 = LayerNorm(h4) in place in B2
    layernorm_kernel<<<N, 256, 0, stream>>>(B2, lng, lnb);

    // 5a) B1 = h5 @ W6[0:2048,:]       (no bias/relu)
    wmma_gemm_f32<2048, 2048, 2048, false, false, false>
        <<<blocks(tilesM, 32), 256, 0, stream>>>(B2, W6, B1, nullptr, tilesM);

    // 5b) h7 = relu(h0 @ W6[2048:2560,:] + b6 + B1)  (in-place accumulate)
    wmma_gemm_f32<512, 2048, 512, true, true, true>
        <<<blocks(tilesM, 32), 256, 0, stream>>>(h0, W6 + (long)2048 * 2048, B1,
                                                 b6, tilesM);

    // 6) out = h7 @ W8 + b8            [N,2048]x[2048,512] -> d_out
    wmma_gemm_f32<2048, 512, 2048, true, false, false>
        <<<blocks(tilesM, 8), 256, 0, stream>>>(B1, W8, (float*)d_out, b8,
                                                tilesM);
}